// QMolGANNoise_8323646620188
// MI455X (gfx1250) — compile-verified
//
#include <hip/hip_runtime.h>

typedef float v2f __attribute__((ext_vector_type(2)));
typedef float v8f __attribute__((ext_vector_type(8)));

#define NQ   8
#define DIM  256
#define NL   3
#define BATCH 65536

// out = a * b  (complex)
#define CMULV(or_, oi_, ar, ai, br, bi)                    \
  do { or_ = (ar) * (br) - (ai) * (bi);                    \
       oi_ = (ar) * (bi) + (ai) * (br); } while (0)

// ---------------------------------------------------------------------------
// Kernel 1: build the popcount-collapsed unitary  Ũ (256 x 12 complex, cols
// 9..11 zero).  Column p = circuit applied to Σ_{popcount(a)=p}|a>.
// 12 waves, one column each; 8 complex amps per lane (a[7:3]=lane, a[2:0]=t).
// ws layout: Re at ws[a*16+p], Im at ws[4096 + a*16+p] (row stride 16 floats).
// ---------------------------------------------------------------------------
__global__ __launch_bounds__(384) void build_u_kernel(const float* __restrict__ weights,
                                                      float* __restrict__ ws) {
  const int lane = threadIdx.x & 31;
  const int p    = threadIdx.x >> 5;       // column 0..11 (popcount class)

  float sr[8], si[8];
#pragma unroll
  for (int t = 0; t < 8; ++t) {
    const int a = (lane << 3) | t;
    sr[t] = (__popc(a) == p) ? 1.0f : 0.0f;   // p>=9 -> all-zero column
    si[t] = 0.0f;
  }

  for (int l = 0; l < NL; ++l) {
    // ---- RY(weights[l][w]) on wire w (acts on bit 7-w) ----
#pragma unroll
    for (int w = 0; w < NQ; ++w) {
      const float th = 0.5f * weights[l * 15 + w];
      const float c = cosf(th), s = sinf(th);
      const int bit = 7 - w;
      if (bit >= 3) {                      // partner amp in another lane
        const int lm = 1 << (bit - 3);
        const float sgn = ((lane >> (bit - 3)) & 1) ? s : -s;
#pragma unroll
        for (int t = 0; t < 8; ++t) {
          const float pr = __shfl_xor(sr[t], lm, 32);
          const float pi = __shfl_xor(si[t], lm, 32);
          sr[t] = c * sr[t] + sgn * pr;    // b=0: c*a0 - s*a1 ; b=1: c*a1 + s*a0
          si[t] = c * si[t] + sgn * pi;
        }
      } else {                             // intra-lane pair
        const int m = 1 << bit;
#pragma unroll
        for (int t = 0; t < 8; ++t) {
          if ((t & m) == 0) {
            const int u = t | m;
            const float a0r = sr[t], a0i = si[t], a1r = sr[u], a1i = si[u];
            sr[t] = c * a0r - s * a1r;  si[t] = c * a0i - s * a1i;
            sr[u] = s * a0r + c * a1r;  si[u] = s * a0i + c * a1i;
          }
        }
      }
    }
    // ---- ZZ(weights[l][8+w]) on wires (w,w+1): diagonal phase ----
#pragma unroll
    for (int w = 0; w < NQ - 1; ++w) {
      const float th = 0.5f * weights[l * 15 + 8 + w];
      const float cz = cosf(th), sz = sinf(th);
      const int b1 = 7 - w, b2 = 6 - w;
#pragma unroll
      for (int t = 0; t < 8; ++t) {
        const int a = (lane << 3) | t;
        const float sg = (((a >> b1) ^ (a >> b2)) & 1) ? sz : -sz;
        const float r0 = sr[t], i0 = si[t];
        sr[t] = cz * r0 - sg * i0;
        si[t] = cz * i0 + sg * r0;
      }
    }
  }

  float* Ur = ws;
  float* Ui = ws + DIM * 16;
#pragma unroll
  for (int t = 0; t < 8; ++t) {
    const int a = (lane << 3) | t;
    Ur[a * 16 + p] = sr[t];
    Ui[a * 16 + p] = si[t];
  }
}

// ---------------------------------------------------------------------------
// Kernel 2: one wave = one tile of 16 batch columns.
// w_p = q0^(8-p) q1^p (9 complex values, K padded to 12);
// state = Ũ(256x12) x W(12x16) via V_WMMA_F32_16X16X4_F32 (complex = 4 real
// accumulations per K-step, -Im(W) pre-negated).  Fused epilogue: |c|^2 and
// 8 sign-weighted sums in registers, shfl_xor(16) cross-half reduce.
// All W values are NAMED scalars (no indexable arrays -> no select trees).
// ---------------------------------------------------------------------------
__global__ __launch_bounds__(256) void qsim_wmma_kernel(const float* __restrict__ z,
                                                        const float* __restrict__ ws,
                                                        float* __restrict__ out) {
  const int lane = threadIdx.x & 31;
  const int half = lane >> 4;                                    // 0 / 1
  const int wid  = blockIdx.x * (blockDim.x >> 5) + (threadIdx.x >> 5);
  const int col  = (wid << 4) + (lane & 15);

  // single-qubit embedding state: q0 = cos(t1/2) e^{-i t2/2}, q1 = sin(t1/2) e^{+i t2/2}
  const float t1 = 0.5f * asinf(z[col * 2 + 0]);
  const float t2 = 0.5f * asinf(z[col * 2 + 1]);
  const float c1 = cosf(t1), s1 = sinf(t1);
  const float c2 = cosf(t2), s2 = sinf(t2);
  const float q0r = c1 * c2, q0i = -c1 * s2;
  const float q1r = s1 * c2, q1i =  s1 * s2;

  // named-scalar powers q0^k (Ak), q1^k (Bk)
  float A2r, A2i, A3r, A3i, A4r, A4i, A5r, A5i, A6r, A6i, A7r, A7i, A8r, A8i;
  float B2r, B2i, B3r, B3i, B4r, B4i, B5r, B5i, B6r, B6i, B7r, B7i, B8r, B8i;
  CMULV(A2r, A2i, q0r, q0i, q0r, q0i);
  CMULV(A3r, A3i, A2r, A2i, q0r, q0i);
  CMULV(A4r, A4i, A3r, A3i, q0r, q0i);
  CMULV(A5r, A5i, A4r, A4i, q0r, q0i);
  CMULV(A6r, A6i, A5r, A5i, q0r, q0i);
  CMULV(A7r, A7i, A6r, A6i, q0r, q0i);
  CMULV(A8r, A8i, A7r, A7i, q0r, q0i);
  CMULV(B2r, B2i, q1r, q1i, q1r, q1i);
  CMULV(B3r, B3i, B2r, B2i, q1r, q1i);
  CMULV(B4r, B4i, B3r, B3i, q1r, q1i);
  CMULV(B5r, B5i, B4r, B4i, q1r, q1i);
  CMULV(B6r, B6i, B5r, B5i, q1r, q1i);
  CMULV(B7r, B7i, B6r, B6i, q1r, q1i);
  CMULV(B8r, B8i, B7r, B7i, q1r, q1i);

  // w_p = q0^(8-p) * q1^p  -- all named scalars
  float w0r = A8r, w0i = A8i;
  float w1r, w1i, w2r, w2i, w3r, w3i, w4r, w4i, w5r, w5i, w6r, w6i, w7r, w7i;
  float w8r = B8r, w8i = B8i;
  CMULV(w1r, w1i, A7r, A7i, q1r, q1i);
  CMULV(w2r, w2i, A6r, A6i, B2r, B2i);
  CMULV(w3r, w3i, A5r, A5i, B3r, B3i);
  CMULV(w4r, w4i, A4r, A4i, B4r, B4i);
  CMULV(w5r, w5i, A3r, A3i, B5r, B5i);
  CMULV(w6r, w6i, A2r, A2i, B6r, B6i);
  CMULV(w7r, w7i, q0r, q0i, B7r, B7i);

  // B operands (16x16x4 f32 layout: lanes 0-15 K rows {0,1}, lanes 16-31 {2,3})
  // K rows 9..11 are zero padding.  18 cndmasks total, loop-invariant.
  const v2f Wr0  = { half ? w2r : w0r,  half ? w3r : w1r };
  const v2f Wi0  = { half ? w2i : w0i,  half ? w3i : w1i };
  const v2f Wni0 = { -Wi0.x, -Wi0.y };
  const v2f Wr1  = { half ? w6r : w4r,  half ? w7r : w5r };
  const v2f Wi1  = { half ? w6i : w4i,  half ? w7i : w5i };
  const v2f Wni1 = { -Wi1.x, -Wi1.y };
  const v2f Wr2  = { half ? 0.f : w8r,  0.f };
  const v2f Wi2  = { half ? 0.f : w8i,  0.f };
  const v2f Wni2 = { -Wi2.x, 0.f };

  const float* Ur = ws + (lane & 15) * 16 + 2 * half;   // A row base (this lane)
  const float* Ui = Ur + DIM * 16;
  const float hsign = half ? -1.f : 1.f;                // bit3 of amplitude index

  float acc0 = 0.f, acc1 = 0.f, acc2 = 0.f, acc3 = 0.f;
  float acc4 = 0.f, acc5 = 0.f, acc6 = 0.f, acc7 = 0.f;

#pragma unroll 4
  for (int m = 0; m < 16; ++m) {
    const float* ur = Ur + m * 256;                     // 16 rows * 16 floats
    const float* ui = Ui + m * 256;
    const v2f Ar0 = *(const v2f*)(ur);
    const v2f Ai0 = *(const v2f*)(ui);
    const v2f Ar1 = *(const v2f*)(ur + 4);
    const v2f Ai1 = *(const v2f*)(ui + 4);
    const v2f Ar2 = *(const v2f*)(ur + 8);
    const v2f Ai2 = *(const v2f*)(ui + 8);

    v8f cr = {0.f,0.f,0.f,0.f,0.f,0.f,0.f,0.f};
    v8f ci = {0.f,0.f,0.f,0.f,0.f,0.f,0.f,0.f};
    // Re += Ur*Wr + Ui*(-Wi) ; Im += Ur*Wi + Ui*Wr
    cr = __builtin_amdgcn_wmma_f32_16x16x4_f32(false, Ar0, false, Wr0,  (short)0, cr, false, false);
    ci = __builtin_amdgcn_wmma_f32_16x16x4_f32(false, Ar0, false, Wi0,  (short)0, ci, false, false);
    cr = __builtin_amdgcn_wmma_f32_16x16x4_f32(false, Ai0, false, Wni0, (short)0, cr, false, false);
    ci = __builtin_amdgcn_wmma_f32_16x16x4_f32(false, Ai0, false, Wr0,  (short)0, ci, false, false);
    cr = __builtin_amdgcn_wmma_f32_16x16x4_f32(false, Ar1, false, Wr1,  (short)0, cr, false, false);
    ci = __builtin_amdgcn_wmma_f32_16x16x4_f32(false, Ar1, false, Wi1,  (short)0, ci, false, false);
    cr = __builtin_amdgcn_wmma_f32_16x16x4_f32(false, Ai1, false, Wni1, (short)0, cr, false, false);
    ci = __builtin_amdgcn_wmma_f32_16x16x4_f32(false, Ai1, false, Wr1,  (short)0, ci, false, false);
    cr = __builtin_amdgcn_wmma_f32_16x16x4_f32(false, Ar2, false, Wr2,  (short)0, cr, false, false);
    ci = __builtin_amdgcn_wmma_f32_16x16x4_f32(false, Ar2, false, Wi2,  (short)0, ci, false, false);
    cr = __builtin_amdgcn_wmma_f32_16x16x4_f32(false, Ai2, false, Wni2, (short)0, cr, false, false);
    ci = __builtin_amdgcn_wmma_f32_16x16x4_f32(false, Ai2, false, Wr2,  (short)0, ci, false, false);

    // amplitude index a = (m<<4) | (half<<3) | r ; out[k] uses bit (7-k)
    const float p0 = cr[0]*cr[0] + ci[0]*ci[0];
    const float p1 = cr[1]*cr[1] + ci[1]*ci[1];
    const float p2 = cr[2]*cr[2] + ci[2]*ci[2];
    const float p3 = cr[3]*cr[3] + ci[3]*ci[3];
    const float p4 = cr[4]*cr[4] + ci[4]*ci[4];
    const float p5 = cr[5]*cr[5] + ci[5]*ci[5];
    const float p6 = cr[6]*cr[6] + ci[6]*ci[6];
    const float p7 = cr[7]*cr[7] + ci[7]*ci[7];
    const float psum = ((p0 + p1) + (p2 + p3)) + ((p4 + p5) + (p6 + p7));

    const float s0 = (m & 8) ? -1.f : 1.f;
    const float s1m = (m & 4) ? -1.f : 1.f;
    const float s2m = (m & 2) ? -1.f : 1.f;
    const float s3m = (m & 1) ? -1.f : 1.f;
    acc0 = fmaf(s0,  psum, acc0);
    acc1 = fmaf(s1m, psum, acc1);
    acc2 = fmaf(s2m, psum, acc2);
    acc3 = fmaf(s3m, psum, acc3);
    acc4 = fmaf(hsign, psum, acc4);
    acc5 += (p0 + p1 + p2 + p3) - (p4 + p5 + p6 + p7);
    acc6 += (p0 + p1 - p2 - p3) + (p4 + p5 - p6 - p7);
    acc7 += (p0 - p1 + p2 - p3) + (p4 - p5 + p6 - p7);
  }

  // lanes l and l^16 hold the same column (M halves) -> one xor-reduce
  acc0 += __shfl_xor(acc0, 16, 32);
  acc1 += __shfl_xor(acc1, 16, 32);
  acc2 += __shfl_xor(acc2, 16, 32);
  acc3 += __shfl_xor(acc3, 16, 32);
  acc4 += __shfl_xor(acc4, 16, 32);
  acc5 += __shfl_xor(acc5, 16, 32);
  acc6 += __shfl_xor(acc6, 16, 32);
  acc7 += __shfl_xor(acc7, 16, 32);

  if (half == 0) {
    float4* dst = (float4*)(out + (size_t)col * 8);
    float4 o0; o0.x = acc0; o0.y = acc1; o0.z = acc2; o0.w = acc3;
    float4 o1; o1.x = acc4; o1.y = acc5; o1.z = acc6; o1.w = acc7;
    dst[0] = o0;
    dst[1] = o1;
  }
}

// ---------------------------------------------------------------------------
extern "C" void kernel_launch(void* const* d_in, const int* in_sizes, int n_in,
                              void* d_out, int out_size, void* d_ws, size_t ws_size,
                              hipStream_t stream) {
  const float* z       = (const float*)d_in[0];   // (65536, 2) f32
  const float* weights = (const float*)d_in[1];   // (3, 15)    f32
  float* out = (float*)d_out;                     // (65536, 8) f32
  float* ws  = (float*)d_ws;                      // needs 32 KB (256*16*2 f32)

  // 12 waves, one Ũ column each (cols 9..11 are zero pads)
  hipLaunchKernelGGL(build_u_kernel, dim3(1), dim3(384), 0, stream, weights, ws);

  // 65536 columns / 16 per wave = 4096 waves; 8 waves (256 thr) per block
  const int blocks = (BATCH / 16) / 8;            // 512
  hipLaunchKernelGGL(qsim_wmma_kernel, dim3(blocks), dim3(256), 0, stream, z, ws, out);
}